// HybridQLSTM_65481071407459
// MI455X (gfx1250) — compile-verified
//
#include <hip/hip_runtime.h>
#include <hip/hip_bf16.h>

#define T_STEPS 64
#define BATCH   1024
#define DIN     1024
#define HID     4
#define NCOL    16          // 4 gates x 4 hidden
#define NSTEP   (DIN / 4)   // 256 wmma K-steps
#define XPITCH  72          // floats per staged row: 288B, 16B-aligned

typedef float v2f __attribute__((ext_vector_type(2)));
typedef float v8f __attribute__((ext_vector_type(8)));

#if __has_builtin(__builtin_amdgcn_global_load_async_to_lds_b128)
#define USE_ASYNC_LDS 1
typedef int v4i_ __attribute__((vector_size(4 * sizeof(int))));
typedef __attribute__((address_space(1))) v4i_* gas_v4i;
typedef __attribute__((address_space(3))) v4i_* las_v4i;
#else
#define USE_ASYNC_LDS 0
#endif

// ---------------------------------------------------------------------------
// Kernel 1: A_pre[T*B, 16] = X[T*B, 1024] @ Wx[1024,16] + bias
// via V_WMMA_F32_16X16X4_F32. 8 waves/block, each wave owns a 16-row M tile.
// X is streamed into wave-private LDS via async global->LDS DMA (ASYNCcnt),
// B fragments for all 256 K-steps are pre-packed in LDS once.
// ---------------------------------------------------------------------------
__global__ __launch_bounds__(256)
void qlstm_gemm_wmma(const float* __restrict__ X,
                     const float* __restrict__ Wf, const float* __restrict__ bf,
                     const float* __restrict__ Wi, const float* __restrict__ bi,
                     const float* __restrict__ Wu, const float* __restrict__ bu,
                     const float* __restrict__ Wo, const float* __restrict__ bo,
                     float* __restrict__ Apre)
{
    // Pre-packed B fragments for all 256 K-steps: [step][lane] -> float2
    __shared__ v2f   WB[NSTEP * 32];          // 64 KB
    __shared__ float XS[8][16][XPITCH];       // per-wave X staging (36.9 KB)

    const int tid  = threadIdx.x;
    const int wave = tid >> 5;
    const int lane = tid & 31;

    // ---- pack B fragments (W is K-major: Wg[k*HID + h], col = g*4 + h) ----
    for (int p = tid; p < NSTEP * 32; p += 256) {
        const int s   = p >> 5;
        const int l   = p & 31;
        const int col = l & 15;
        const int g   = col >> 2;
        const int h   = col & 3;
        const int k   = s * 4 + ((l >> 4) << 1);   // K offset per lane half
        const float* W = (g == 0) ? Wf : (g == 1) ? Wi : (g == 2) ? Wu : Wo;
        v2f v;
        v.x = W[(k    ) * HID + h];
        v.y = W[(k + 1) * HID + h];
        WB[p] = v;
    }
    __syncthreads();

    const int   col  = lane & 15;
    const int   half = lane >> 4;
    const int   gb   = col >> 2;
    const float bias = ((gb == 0) ? bf : (gb == 1) ? bi : (gb == 2) ? bu : bo)[col & 3];

    const long Mbase = (long)(blockIdx.x * 8 + wave) * 16;

    v8f acc = {};
    for (int kb = 0; kb < DIN; kb += 64) {
        // ---- stage 16 rows x 64 cols of X into this wave's LDS slab ----
#if USE_ASYNC_LDS
        // WAR: previous chunk's ds reads must drain before DMA overwrites XS
        asm volatile("s_wait_dscnt 0x0" ::: "memory");
        #pragma unroll
        for (int j = 0; j < 8; ++j) {
            const int p  = j * 32 + lane;          // 256 x 16B = 4KB chunk
            const int r  = p >> 4;
            const int c4 = (p & 15) * 4;
            const float* gp = X + (Mbase + r) * (long)DIN + kb + c4;
            __builtin_amdgcn_global_load_async_to_lds_b128(
                (gas_v4i)gp, (las_v4i)&XS[wave][r][c4], 0, 0);
        }
        asm volatile("s_wait_asynccnt 0x0" ::: "memory");
#else
        #pragma unroll
        for (int j = 0; j < 8; ++j) {
            const int p  = j * 32 + lane;
            const int r  = p >> 4;
            const int c4 = (p & 15) * 4;
            const float4 v = *(const float4*)(X + (Mbase + r) * (long)DIN + kb + c4);
            *(float4*)&XS[wave][r][c4] = v;
        }
        __syncthreads();
#endif
        // ---- 16 WMMA K-steps over the staged chunk ----
        #pragma unroll
        for (int i = 0; i < 16; ++i) {
            const int kk = i * 4 + half * 2;      // local K per lane half
            v2f a;
            a.x = XS[wave][lane & 15][kk];
            a.y = XS[wave][lane & 15][kk + 1];
            const v2f b = WB[((kb >> 2) + i) * 32 + lane];
            acc = __builtin_amdgcn_wmma_f32_16x16x4_f32(
                      false, a, false, b, (short)0, acc, false, false);
        }
#if !USE_ASYNC_LDS
        __syncthreads();
#endif
    }

    // C/D layout: VGPR j -> row half*8 + j, col = lane&15
    #pragma unroll
    for (int j = 0; j < 8; ++j) {
        const long row = Mbase + half * 8 + j;
        Apre[row * 16 + col] = acc[j] + bias;
    }
}

// ---------------------------------------------------------------------------
// Closed-form qlayer gate:  m_h = cos(y_h + th_h),
// z = {m1m2m3, m0m1, m0m1m2, m0m1m2m3},  out = sigmoid(z + k)
// ---------------------------------------------------------------------------
__device__ __forceinline__ float sigf(float x) { return 1.f / (1.f + expf(-x)); }

__device__ __forceinline__ void qgate(const float* y, const float* th, float k,
                                      float* out)
{
    float m0 = cosf(y[0] + th[0]);
    float m1 = cosf(y[1] + th[1]);
    float m2 = cosf(y[2] + th[2]);
    float m3 = cosf(y[3] + th[3]);
    float z1 = m0 * m1;
    float z2 = z1 * m2;
    out[0] = sigf(m1 * m2 * m3 + k);
    out[1] = sigf(z1 + k);
    out[2] = sigf(z2 + k);
    out[3] = sigf(z2 * m3 + k);
}

// ---------------------------------------------------------------------------
// Kernel 2: simulate batch row 0 sequentially to extract k[t] (the only
// cross-batch coupling in the whole recurrence).
// ---------------------------------------------------------------------------
__global__ void qlstm_row0(const float* __restrict__ X,
                           const float* __restrict__ Apre,
                           const float* __restrict__ Wf, const float* __restrict__ Wi,
                           const float* __restrict__ Wu, const float* __restrict__ Wo,
                           const float* __restrict__ thf, const float* __restrict__ thi,
                           const float* __restrict__ tho,
                           float* __restrict__ karr)
{
    if (threadIdx.x != 0) return;
    float hx[4] = {0, 0, 0, 0}, cx[4] = {0, 0, 0, 0};
    float Wh[4][16];
    for (int r = 0; r < 4; ++r)
        for (int c = 0; c < 16; ++c) {
            const int g = c >> 2, h = c & 3;
            const float* W = (g == 0) ? Wf : (g == 1) ? Wi : (g == 2) ? Wu : Wo;
            Wh[r][c] = W[(DIN + r) * HID + h];
        }
    float th[3][4];
    for (int h = 0; h < 4; ++h) { th[0][h] = thf[h]; th[1][h] = thi[h]; th[2][h] = tho[h]; }

    for (int t = 0; t < T_STEPS; ++t) {
        float kv = 1.f;
        for (int i = 0; i < 4; ++i)
            kv *= cosf((X[(long)t * BATCH * DIN + i] - hx[i]) * 0.5f);
        kv = fabsf(kv);
        karr[t] = kv;

        float y[16];
        for (int c = 0; c < 16; ++c) {
            float v = Apre[((long)t * BATCH) * 16 + c];
            for (int r = 0; r < 4; ++r) v += hx[r] * Wh[r][c];
            y[c] = v;
        }
        float f4[4], i4[4], o4[4];
        qgate(&y[0],  th[0], kv, f4);
        qgate(&y[4],  th[1], kv, i4);
        qgate(&y[12], th[2], kv, o4);
        for (int h = 0; h < 4; ++h) {
            const float g4 = tanhf(y[8 + h]);
            cx[h] = f4[h] * cx[h] + i4[h] * g4;
            hx[h] = o4[h] * tanhf(cx[h]);
        }
    }
}

// ---------------------------------------------------------------------------
// Kernel 3: with k[t] known, every batch row's recurrence is independent.
// One thread per batch row; Wh/theta/k broadcast from LDS.
// ---------------------------------------------------------------------------
__global__ __launch_bounds__(256)
void qlstm_scan(const float* __restrict__ Apre, const float* __restrict__ karr,
                const float* __restrict__ Wf, const float* __restrict__ Wi,
                const float* __restrict__ Wu, const float* __restrict__ Wo,
                const float* __restrict__ thf, const float* __restrict__ thi,
                const float* __restrict__ tho,
                float* __restrict__ out)
{
    __shared__ float Wh[4][16];
    __shared__ float kS[T_STEPS];
    __shared__ float thS[3][4];

    const int tid = threadIdx.x;
    if (tid < 64) {
        const int r = tid >> 4, c = tid & 15;
        const int g = c >> 2, h = c & 3;
        const float* W = (g == 0) ? Wf : (g == 1) ? Wi : (g == 2) ? Wu : Wo;
        Wh[r][c] = W[(DIN + r) * HID + h];
    }
    if (tid < T_STEPS) kS[tid] = karr[tid];
    if (tid < 12) {
        const int g = tid >> 2, h = tid & 3;
        thS[g][h] = ((g == 0) ? thf : (g == 1) ? thi : tho)[h];
    }
    __syncthreads();

    const int b = blockIdx.x * 256 + tid;
    float hx[4] = {0, 0, 0, 0}, cx[4] = {0, 0, 0, 0};

    for (int t = 0; t < T_STEPS; ++t) {
        const float* ap = Apre + ((long)t * BATCH + b) * 16;
        float y[16];
        #pragma unroll
        for (int c = 0; c < 16; c += 4) {
            const float4 v = *(const float4*)(ap + c);
            y[c] = v.x; y[c + 1] = v.y; y[c + 2] = v.z; y[c + 3] = v.w;
        }
        #pragma unroll
        for (int c = 0; c < 16; ++c) {
            float v = y[c];
            #pragma unroll
            for (int r = 0; r < 4; ++r) v += hx[r] * Wh[r][c];
            y[c] = v;
        }
        const float kv = kS[t];
        float f4[4], i4[4], o4[4];
        qgate(&y[0],  thS[0], kv, f4);
        qgate(&y[4],  thS[1], kv, i4);
        qgate(&y[12], thS[2], kv, o4);
        #pragma unroll
        for (int h = 0; h < 4; ++h) {
            const float g4 = tanhf(y[8 + h]);
            cx[h] = f4[h] * cx[h] + i4[h] * g4;
            hx[h] = o4[h] * tanhf(cx[h]);
        }
        const float4 hv = {hx[0], hx[1], hx[2], hx[3]};
        *(float4*)(out + ((long)t * BATCH + b) * 4) = hv;
    }
    const float4 hv = {hx[0], hx[1], hx[2], hx[3]};
    const float4 cv = {cx[0], cx[1], cx[2], cx[3]};
    *(float4*)(out + (long)T_STEPS * BATCH * 4 + b * 4) = hv;
    *(float4*)(out + (long)T_STEPS * BATCH * 4 + (long)BATCH * 4 + b * 4) = cv;
}

// ---------------------------------------------------------------------------
extern "C" void kernel_launch(void* const* d_in, const int* in_sizes, int n_in,
                              void* d_out, int out_size, void* d_ws, size_t ws_size,
                              hipStream_t stream)
{
    (void)in_sizes; (void)n_in; (void)out_size; (void)ws_size;

    const float* X   = (const float*)d_in[0];
    const float* Wf  = (const float*)d_in[1];
    const float* bf  = (const float*)d_in[2];
    const float* Wi  = (const float*)d_in[3];
    const float* bi  = (const float*)d_in[4];
    const float* Wu  = (const float*)d_in[5];
    const float* bu  = (const float*)d_in[6];
    const float* Wo  = (const float*)d_in[7];
    const float* bo  = (const float*)d_in[8];
    const float* thf = (const float*)d_in[9];
    const float* thi = (const float*)d_in[10];
    const float* thu = (const float*)d_in[11]; (void)thu;  // unused: u-gate has no qlayer
    const float* tho = (const float*)d_in[12];

    float* out  = (float*)d_out;
    float* Apre = (float*)d_ws;                                   // T*B*16 floats = 4 MB
    float* karr = Apre + (size_t)T_STEPS * BATCH * 16;            // 64 floats

    // (T*B)/16 M-tiles, 8 waves per block -> 512 blocks
    qlstm_gemm_wmma<<<(T_STEPS * BATCH) / (8 * 16), 256, 0, stream>>>(
        X, Wf, bf, Wi, bi, Wu, bu, Wo, bo, Apre);

    qlstm_row0<<<1, 32, 0, stream>>>(X, Apre, Wf, Wi, Wu, Wo, thf, thi, tho, karr);

    qlstm_scan<<<BATCH / 256, 256, 0, stream>>>(
        Apre, karr, Wf, Wi, Wu, Wo, thf, thi, tho, out);
}